// GCN_38603166056515
// MI455X (gfx1250) — compile-verified
//
#include <hip/hip_runtime.h>

typedef float v2f __attribute__((ext_vector_type(2)));
typedef float v8f __attribute__((ext_vector_type(8)));

#define D 64  // D_IN == HIDDEN == 64

// ---------------------------------------------------------------------------
// Degree / normalization kernels
// ---------------------------------------------------------------------------
__global__ void k_deg_init(float* __restrict__ deg, int n) {
    int i = blockIdx.x * blockDim.x + threadIdx.x;
    if (i < n) deg[i] = 1.0f;  // self-loop contributes 1 to every node's degree
}

__global__ void k_deg_edges(float* __restrict__ deg, const int* __restrict__ dst,
                            int n_edges) {
    int e = blockIdx.x * blockDim.x + threadIdx.x;
    if (e < n_edges) atomicAdd(&deg[dst[e]], 1.0f);
}

__global__ void k_deg_to_dinv(float* __restrict__ deg, int n) {
    int i = blockIdx.x * blockDim.x + threadIdx.x;
    if (i < n) deg[i] = rsqrtf(deg[i]);  // deg >= 1 always (self loop)
}

// ---------------------------------------------------------------------------
// Fused (optional bias+ReLU) + GEMM via V_WMMA_F32_16X16X4_F32
// out[N x 64] = act(in[N x 64]) @ W[64 x 64]
// act(x) = HAS_BIAS ? max(x + bias[k], 0) : x   (bias indexed by input feat k)
// Block = 128 threads (4 waves). Each block: one 16-row tile of output.
// Wave w computes columns [16w, 16w+16).  K loop: 16 steps of 4.
// HAS_BIAS is a template parameter so the unrolled WMMA loop is branch-free.
// ---------------------------------------------------------------------------
template <bool HAS_BIAS>
__global__ __launch_bounds__(128) void k_gemm_wmma(
    const float* __restrict__ in, const float* __restrict__ W,
    const float* __restrict__ bias, float* __restrict__ out, int n_nodes)
{
    const int tile  = blockIdx.x;
    const int lane  = threadIdx.x & 31;
    const int wv    = threadIdx.x >> 5;          // 0..3: column tile
    const int m     = lane & 15;
    const int khalf = (lane >> 4) ? 2 : 0;       // A/B K sub-offset per half-wave
    int row = tile * 16 + m;
    if (row >= n_nodes) row = n_nodes - 1;       // keep EXEC all-1s for WMMA
    const int col = wv * 16 + m;

    const float* __restrict__ arow = in + (size_t)row * D;
    v8f c = {};
#pragma unroll
    for (int k0 = 0; k0 < D; k0 += 4) {
        const int ka = k0 + khalf;
        v2f a, b;
        a.x = arow[ka];
        a.y = arow[ka + 1];
        if (HAS_BIAS) {
            a.x = fmaxf(a.x + bias[ka], 0.0f);
            a.y = fmaxf(a.y + bias[ka + 1], 0.0f);
        }
        b.x = W[(size_t)ka * D + col];
        b.y = W[(size_t)(ka + 1) * D + col];
        // D = A(16x4) * B(4x16) + C   -- exact fp32 MMA
        c = __builtin_amdgcn_wmma_f32_16x16x4_f32(
                /*neg_a=*/false, a, /*neg_b=*/false, b,
                /*c_mod=*/(short)0, c, /*reuse_a=*/false, /*reuse_b=*/false);
    }

    // C/D layout: VGPR r -> M = r + (lane<16 ? 0 : 8), N = lane&15
    const int rbase = tile * 16 + ((lane >> 4) ? 8 : 0);
#pragma unroll
    for (int r = 0; r < 8; ++r) {
        const int orow = rbase + r;
        if (orow < n_nodes) out[(size_t)orow * D + col] = c[r];
    }
}

// ---------------------------------------------------------------------------
// Self-loop initialization: out[n,:] = in[n,:] * dinv[n]^2
// 16 threads per node, float4 each.
// ---------------------------------------------------------------------------
__global__ void k_self_init(const float* __restrict__ in, float* __restrict__ out,
                            const float* __restrict__ dinv, int n_nodes)
{
    int gid = blockIdx.x * blockDim.x + threadIdx.x;
    int n = gid >> 4, q = gid & 15;
    if (n >= n_nodes) return;
    float w = dinv[n]; w *= w;
    const float4 v = *(const float4*)(in + (size_t)n * D + q * 4);
    float4 r; r.x = v.x * w; r.y = v.y * w; r.z = v.z * w; r.w = v.w * w;
    *(float4*)(out + (size_t)n * D + q * 4) = r;
}

// ---------------------------------------------------------------------------
// Edge scatter: out[dst,:] += in[src,:] * dinv[src]*dinv[dst]
// 16 threads per edge; each thread handles a float4 slice (L2-resident atomics).
// All index math kept in 32-bit (E*16 = 25.6M fits fine).
// ---------------------------------------------------------------------------
__global__ void k_edge_agg(const float* __restrict__ in, float* __restrict__ out,
                           const int* __restrict__ src, const int* __restrict__ dst,
                           const float* __restrict__ dinv, int n_edges)
{
    const int gid = blockIdx.x * blockDim.x + threadIdx.x;
    const int e = gid >> 4;
    const int q = gid & 15;
    if (e >= n_edges) return;
    const int s = src[e], d = dst[e];
    const float norm = dinv[s] * dinv[d];
    const float4 v = *(const float4*)(in + (size_t)s * D + q * 4);
    float* o = out + (size_t)d * D + q * 4;
    atomicAdd(o + 0, v.x * norm);
    atomicAdd(o + 1, v.y * norm);
    atomicAdd(o + 2, v.z * norm);
    atomicAdd(o + 3, v.w * norm);
}

// ---------------------------------------------------------------------------
// Readout: out[n] = relu(agg[n,:] + b2) . Wo + bo
// ---------------------------------------------------------------------------
__global__ void k_readout(const float* __restrict__ agg, const float* __restrict__ b2,
                          const float* __restrict__ Wo, const float* __restrict__ bo,
                          float* __restrict__ out, int n_nodes)
{
    int n = blockIdx.x * blockDim.x + threadIdx.x;
    if (n >= n_nodes) return;
    const float* r = agg + (size_t)n * D;
    float acc = bo[0];
#pragma unroll
    for (int k = 0; k < D; k += 4) {
        const float4 v = *(const float4*)(r + k);
        acc += fmaxf(v.x + b2[k + 0], 0.0f) * Wo[k + 0];
        acc += fmaxf(v.y + b2[k + 1], 0.0f) * Wo[k + 1];
        acc += fmaxf(v.z + b2[k + 2], 0.0f) * Wo[k + 2];
        acc += fmaxf(v.w + b2[k + 3], 0.0f) * Wo[k + 3];
    }
    out[n] = acc;
}

// ---------------------------------------------------------------------------
extern "C" void kernel_launch(void* const* d_in, const int* in_sizes, int n_in,
                              void* d_out, int out_size, void* d_ws, size_t ws_size,
                              hipStream_t stream)
{
    const float* h  = (const float*)d_in[0];
    const int*   ei = (const int*)d_in[1];   // [2, E] flattened: src then dst
    const float* W1 = (const float*)d_in[2];
    const float* b1 = (const float*)d_in[3];
    const float* W2 = (const float*)d_in[4];
    const float* b2 = (const float*)d_in[5];
    const float* Wo = (const float*)d_in[6];
    const float* bo = (const float*)d_in[7];

    const int N = in_sizes[0] / D;
    const int E = in_sizes[1] / 2;
    const int* src = ei;
    const int* dst = ei + E;

    // workspace layout: dinv[N] | bufA[N*D] | bufB[N*D]
    float* dinv = (float*)d_ws;
    float* bufA = dinv + N;
    float* bufB = bufA + (size_t)N * D;
    float* outp = (float*)d_out;

    const int BT = 256;
    const int gN     = (N + BT - 1) / BT;
    const int gN16   = (int)(((long long)N * 16 + BT - 1) / BT);
    const int gE     = (E + BT - 1) / BT;
    const int gE16   = (int)(((long long)E * 16 + BT - 1) / BT);
    const int gTiles = (N + 15) / 16;

    // --- symmetric normalization: dinv = rsqrt(indeg + 1) ---
    k_deg_init<<<gN, BT, 0, stream>>>(dinv, N);
    k_deg_edges<<<gE, BT, 0, stream>>>(dinv, dst, E);
    k_deg_to_dinv<<<gN, BT, 0, stream>>>(dinv, N);

    // --- layer 1: h1 = h @ W1; agg1 = scatter(h1) ---
    k_gemm_wmma<false><<<gTiles, 128, 0, stream>>>(h, W1, nullptr, bufA, N);
    k_self_init<<<gN16, BT, 0, stream>>>(bufA, bufB, dinv, N);
    k_edge_agg<<<gE16, BT, 0, stream>>>(bufA, bufB, src, dst, dinv, E);

    // --- layer 2: h2 = relu(agg1 + b1) @ W2; agg2 = scatter(h2) ---
    k_gemm_wmma<true><<<gTiles, 128, 0, stream>>>(bufB, W2, b1, bufA, N);
    k_self_init<<<gN16, BT, 0, stream>>>(bufA, bufB, dinv, N);
    k_edge_agg<<<gE16, BT, 0, stream>>>(bufA, bufB, src, dst, dinv, E);

    // --- readout: out = relu(agg2 + b2) @ Wo + bo ---
    k_readout<<<gN, BT, 0, stream>>>(bufB, b2, Wo, bo, outp, N);
}